// SynGNNLayer_70308614636216
// MI455X (gfx1250) — compile-verified
//
#include <hip/hip_runtime.h>
#include <hip/hip_bf16.h>

typedef float v2f __attribute__((ext_vector_type(2)));
typedef float v8f __attribute__((ext_vector_type(8)));

#define NNODE 50000
#define NEDGE 400000
#define DD    128
#define HH    4
#define CC    128
#define DEE   64
#define FFD   2048
#define HC    512   // H*C
#define KC    64    // GEMM K-chunk staged through LDS

// ---------------------------------------------------------------------------
// helpers
// ---------------------------------------------------------------------------
__device__ inline void atomicMaxF(float* addr, float val) {
    unsigned int* a = (unsigned int*)addr;
    unsigned int old = __float_as_uint(*addr);
    while (__uint_as_float(old) < val) {
        unsigned int assumed = old;
        old = atomicCAS(a, assumed, __float_as_uint(val));
        if (old == assumed) break;
    }
}

// ---------------------------------------------------------------------------
// init workspace accumulators
// ---------------------------------------------------------------------------
__global__ void __launch_bounds__(256)
k_init(float* scal, float* mx, float* den, float* agg, int nh, long long nd) {
    long long i = (long long)blockIdx.x * blockDim.x + threadIdx.x;
    long long stride = (long long)gridDim.x * blockDim.x;
    for (; i < nd; i += stride) {
        if (i < 16)  scal[i] = 0.0f;
        if (i < nh) { mx[i] = -3.402823466e38f; den[i] = 0.0f; }
        agg[i] = 0.0f;
    }
}

// ---------------------------------------------------------------------------
// grid-stride sum / sum-of-squares reduction -> scalar slots
// ---------------------------------------------------------------------------
__global__ void __launch_bounds__(256)
k_reduce(const float* __restrict__ x, long long n, float* outS, float* outQ) {
    float s = 0.0f, q = 0.0f;
    long long i = (long long)blockIdx.x * blockDim.x + threadIdx.x;
    long long stride = (long long)gridDim.x * blockDim.x;
    for (; i < n; i += stride) { float v = x[i]; s += v; q += v * v; }
    for (int m = 16; m > 0; m >>= 1) { s += __shfl_xor(s, m, 32); q += __shfl_xor(q, m, 32); }
    __shared__ float ss[8], sq[8];
    int wave = threadIdx.x >> 5, lane = threadIdx.x & 31;
    if (lane == 0) { ss[wave] = s; sq[wave] = q; }
    __syncthreads();
    if (threadIdx.x == 0) {
        float S = 0.0f, Q = 0.0f;
        for (int w = 0; w < 8; ++w) { S += ss[w]; Q += sq[w]; }
        atomicAdd(outS, S); atomicAdd(outQ, Q);
    }
}

// ---------------------------------------------------------------------------
// graph layernorm apply: out = (x - mean)/(sqrt(var)+eps) * w[d] + b[d]
// ---------------------------------------------------------------------------
__global__ void __launch_bounds__(256)
k_ln_apply(const float* __restrict__ x, const float* __restrict__ sums,
           const float* __restrict__ w, const float* __restrict__ b,
           float* __restrict__ out, long long n, float invN) {
    float mean = sums[0] * invN;
    float var  = sums[1] * invN - mean * mean;
    float inv  = 1.0f / (sqrtf(fmaxf(var, 0.0f)) + 1e-5f);
    long long i = (long long)blockIdx.x * blockDim.x + threadIdx.x;
    long long stride = (long long)gridDim.x * blockDim.x;
    for (; i < n; i += stride) {
        int d = (int)(i & (DD - 1));
        out[i] = (x[i] - mean) * inv * w[d] + b[d];
    }
}

// ---------------------------------------------------------------------------
// fp32 WMMA GEMM. Block = 8 waves sharing one 64-row M-block; wave w owns
// N-tile (ngroup*8 + w). 64xKC A-panel staged through LDS with register
// prefetch of the next chunk; B fragments loaded from global (L2-resident).
// C = act(A[M,K] @ B[K,Nn] + bias) (+ residual)
// grid.x = mblks * (Nn/128); requires Nn % 128 == 0, K % KC == 0.
// ---------------------------------------------------------------------------
__global__ void __launch_bounds__(256)
k_gemm_wmma(const float* __restrict__ A, const float* __restrict__ B,
            const float* __restrict__ bias, const float* __restrict__ resid,
            float* __restrict__ C, int M, int K, int Nn, int act) {
    __shared__ float sAt[64 * KC];                 // 16 KB
    const int t    = threadIdx.x;
    const int lane = t & 31;
    const int wave = t >> 5;
    const int nGroups = Nn >> 7;                   // (Nn/16)/8
    const int mblk = blockIdx.x / nGroups;
    const int ng   = blockIdx.x - mblk * nGroups;
    const int m0 = mblk * 64;
    const int n0 = (ng * 8 + wave) << 4;
    const int row  = lane & 15;
    const int half = lane >> 4;
    const int kA   = half * 2;

    // register-staged A chunk: 4 x float4 per thread (64 rows x KC floats)
    float4 st[4];
    const int gr[4] = { (t + 0)   >> 4, (t + 256) >> 4, (t + 512) >> 4, (t + 768) >> 4 };
    const int gk[4] = { (t & 15) * 4, (t & 15) * 4, (t & 15) * 4, (t & 15) * 4 };
#pragma unroll
    for (int u = 0; u < 4; ++u)
        st[u] = *(const float4*)(A + (size_t)(m0 + gr[u]) * K + 0 + gk[u]);

    v8f c0 = {}, c1 = {}, c2 = {}, c3 = {};
    const float* bp0 = B + (size_t)kA * Nn + n0 + row;

    for (int kc = 0; kc < K; kc += KC) {
        // commit staged chunk to LDS
#pragma unroll
        for (int u = 0; u < 4; ++u)
            *(float4*)(&sAt[gr[u] * KC + gk[u]]) = st[u];
        __syncthreads();
        // prefetch next chunk into registers while computing this one
        if (kc + KC < K) {                         // wave-uniform
#pragma unroll
            for (int u = 0; u < 4; ++u)
                st[u] = *(const float4*)(A + (size_t)(m0 + gr[u]) * K + (kc + KC) + gk[u]);
        }
        const float* bp = bp0 + (size_t)kc * Nn;
#pragma unroll
        for (int kb = 0; kb < KC; kb += 4) {
            v2f b;
            b.x = bp[0]; b.y = bp[Nn];
            v2f a0v = *(const v2f*)(&sAt[(row)      * KC + kb + kA]);
            v2f a1v = *(const v2f*)(&sAt[(16 + row) * KC + kb + kA]);
            v2f a2v = *(const v2f*)(&sAt[(32 + row) * KC + kb + kA]);
            v2f a3v = *(const v2f*)(&sAt[(48 + row) * KC + kb + kA]);
            c0 = __builtin_amdgcn_wmma_f32_16x16x4_f32(false, a0v, false, b, (short)0, c0, false, false);
            c1 = __builtin_amdgcn_wmma_f32_16x16x4_f32(false, a1v, false, b, (short)0, c1, false, false);
            c2 = __builtin_amdgcn_wmma_f32_16x16x4_f32(false, a2v, false, b, (short)0, c2, false, false);
            c3 = __builtin_amdgcn_wmma_f32_16x16x4_f32(false, a3v, false, b, (short)0, c3, false, false);
            bp += (size_t)4 * Nn;
        }
        __syncthreads();
    }

    const int col = lane & 15;
    const float bv = bias ? bias[n0 + col] : 0.0f;
#pragma unroll
    for (int sub = 0; sub < 4; ++sub) {
        const int m1 = m0 + sub * 16;
        if (m1 >= M) break;                        // wave-uniform (M % 16 == 0)
        v8f c = (sub == 0) ? c0 : (sub == 1) ? c1 : (sub == 2) ? c2 : c3;
#pragma unroll
        for (int i = 0; i < 8; ++i) {
            int r = half * 8 + i;
            float v = c[i] + bv;
            if (act == 1) v = 0.5f * v * (1.0f + erff(v * 0.70710678f)); // exact GELU
            if (resid) v += resid[(size_t)(m1 + r) * Nn + n0 + col];
            C[(size_t)(m1 + r) * Nn + n0 + col] = v;
        }
    }
}

// ---------------------------------------------------------------------------
// fused edge kernel: eproj tile (16 edges x 512) via WMMA, + gathered xl/xr,
// leaky_relu, att dot -> logits[E,H]; atomic segment max into mx[N,H]
// ---------------------------------------------------------------------------
__global__ void __launch_bounds__(256)
k_edge_logits(const float* __restrict__ edge_attr, const float* __restrict__ W_e,
              const float* __restrict__ att, const int* __restrict__ edge_index,
              const float* __restrict__ xl, const float* __restrict__ xr,
              float* __restrict__ logits, float* __restrict__ mx, int ntiles) {
    const int lane = threadIdx.x & 31;
    const int wave = threadIdx.x >> 5;
    const int tile = blockIdx.x * 8 + wave;
    if (tile >= ntiles) return;                  // wave-uniform
    const int e0   = tile << 4;
    const int row  = lane & 15;
    const int half = lane >> 4;
    const int kA   = half * 2;
    v2f aFr[16];
#pragma unroll
    for (int kk = 0; kk < 16; ++kk) {
        const float* ap = edge_attr + (size_t)(e0 + row) * DEE + kk * 4 + kA;
        aFr[kk].x = ap[0]; aFr[kk].y = ap[1];
    }
    const int rbase = half * 8;
    int srcs[8], dsts[8];
#pragma unroll
    for (int i = 0; i < 8; ++i) {
        int ei = e0 + rbase + i;
        srcs[i] = edge_index[ei];
        dsts[i] = edge_index[NEDGE + ei];
    }
    float acc[8][HH];
#pragma unroll
    for (int i = 0; i < 8; ++i)
#pragma unroll
        for (int h = 0; h < HH; ++h) acc[i][h] = 0.0f;

    const int col = lane & 15;
    for (int nb = 0; nb < 32; ++nb) {            // 32 n-blocks cover H*C = 512
        const int n0 = nb << 4;
        v8f c = {};
#pragma unroll
        for (int kk = 0; kk < 16; ++kk) {
            v2f b;
            const float* bp = W_e + (size_t)(kk * 4 + kA) * HC + n0 + col;
            b.x = bp[0]; b.y = bp[HC];
            c = __builtin_amdgcn_wmma_f32_16x16x4_f32(false, aFr[kk], false, b,
                                                      (short)0, c, false, false);
        }
        const int hc = n0 + col;                 // each n-block sits in one head
        const int h  = hc >> 7;
        const float av = att[hc];
#pragma unroll
        for (int i = 0; i < 8; ++i) {
            float v = c[i] + xl[(size_t)srcs[i] * HC + hc]
                           + xr[(size_t)dsts[i] * HC + hc];
            v = v > 0.0f ? v : 0.2f * v;         // leaky_relu(0.2)
            acc[i][h] += av * v;
        }
    }
#pragma unroll
    for (int m = 1; m < 16; m <<= 1)
#pragma unroll
        for (int i = 0; i < 8; ++i)
#pragma unroll
            for (int h = 0; h < HH; ++h)
                acc[i][h] += __shfl_xor(acc[i][h], m, 32);
    if (col == 0) {                              // lanes 0 and 16 commit
#pragma unroll
        for (int i = 0; i < 8; ++i)
#pragma unroll
            for (int h = 0; h < HH; ++h) {
                logits[(size_t)(e0 + rbase + i) * HH + h] = acc[i][h];
                atomicMaxF(&mx[(size_t)dsts[i] * HH + h], acc[i][h]);
            }
    }
}

// ---------------------------------------------------------------------------
// segment softmax: exp(logit - segmax) in place; atomic denom accumulation
// ---------------------------------------------------------------------------
__global__ void __launch_bounds__(256)
k_edge_exp(const int* __restrict__ edge_index, float* __restrict__ logits,
           const float* __restrict__ mx, float* __restrict__ den) {
    int i = blockIdx.x * blockDim.x + threadIdx.x;
    if (i >= NEDGE * HH) return;
    int e = i >> 2, h = i & 3;
    int d = edge_index[NEDGE + e];
    float ex = __expf(logits[i] - mx[(size_t)d * HH + h]);
    logits[i] = ex;
    atomicAdd(&den[(size_t)d * HH + h], ex);
}

// ---------------------------------------------------------------------------
// alpha = ex/(den+1e-16); write alpha to out; head-reduce in registers then
// scatter 128 channels/edge. one wave per edge.
// ---------------------------------------------------------------------------
__global__ void __launch_bounds__(256)
k_edge_agg(const float* __restrict__ ex, const float* __restrict__ den,
           const int* __restrict__ edge_index, const float* __restrict__ xl,
           float* __restrict__ agg, float* __restrict__ alpha_out) {
    int wv   = (blockIdx.x * blockDim.x + threadIdx.x) >> 5;
    int lane = threadIdx.x & 31;
    if (wv >= NEDGE) return;
    int s = edge_index[wv], d = edge_index[NEDGE + wv];
    float al[HH];
#pragma unroll
    for (int h = 0; h < HH; ++h)
        al[h] = ex[(size_t)wv * HH + h] / (den[(size_t)d * HH + h] + 1e-16f);
    if (lane < HH) alpha_out[(size_t)wv * HH + lane] = al[lane];
#pragma unroll
    for (int j = 0; j < 4; ++j) {
        int c = lane + 32 * j;
        float v = 0.0f;
#pragma unroll
        for (int h = 0; h < HH; ++h)
            v += al[h] * xl[(size_t)s * HC + h * CC + c];
        atomicAdd(&agg[(size_t)d * CC + c], 0.25f * v);
    }
}

// ---------------------------------------------------------------------------
// srcB = srcA + agg + bias_out  (into agg)
// ---------------------------------------------------------------------------
__global__ void __launch_bounds__(256)
k_res_bias(const float* __restrict__ srcA, const float* __restrict__ bias,
           float* __restrict__ agg, long long n) {
    long long i = (long long)blockIdx.x * blockDim.x + threadIdx.x;
    long long stride = (long long)gridDim.x * blockDim.x;
    for (; i < n; i += stride)
        agg[i] = srcA[i] + agg[i] + bias[i & (DD - 1)];
}

// ---------------------------------------------------------------------------
// cast edge_index int32 -> float into output tuple slot
// ---------------------------------------------------------------------------
__global__ void __launch_bounds__(256)
k_cast_idx(const int* __restrict__ idx, float* __restrict__ out, int n) {
    int i = blockIdx.x * blockDim.x + threadIdx.x;
    if (i < n) out[i] = (float)idx[i];
}

// ---------------------------------------------------------------------------
extern "C" void kernel_launch(void* const* d_in, const int* in_sizes, int n_in,
                              void* d_out, int out_size, void* d_ws, size_t ws_size,
                              hipStream_t stream) {
    const float* x        = (const float*)d_in[0];
    const int*   eidx     = (const int*)d_in[1];
    const float* eattr    = (const float*)d_in[2];
    const float* W_l      = (const float*)d_in[3];
    const float* b_l      = (const float*)d_in[4];
    const float* W_r      = (const float*)d_in[5];
    const float* b_r      = (const float*)d_in[6];
    const float* W_e      = (const float*)d_in[7];
    const float* att      = (const float*)d_in[8];
    const float* bias_out = (const float*)d_in[9];
    const float* ln0_w = (const float*)d_in[10], *ln0_b = (const float*)d_in[11];
    const float* ln1_w = (const float*)d_in[12], *ln1_b = (const float*)d_in[13];
    const float* ln2_w = (const float*)d_in[14], *ln2_b = (const float*)d_in[15];
    const float* W1 = (const float*)d_in[16], *b1 = (const float*)d_in[17];
    const float* W2 = (const float*)d_in[18], *b2 = (const float*)d_in[19];

    const long long ND = (long long)NNODE * DD;        // 6.4M
    float* W    = (float*)d_ws;
    float* scal = W;                                   // 16 scalar slots
    float* sA   = scal + 16;                           // N*D
    float* xl   = sA  + ND;                            // N*512
    float* xr   = xl  + (long long)NNODE * HC;         // N*512
    float* lg   = xr  + (long long)NNODE * HC;         // E*H (logits -> exp)
    float* mx   = lg  + (long long)NEDGE * HH;         // N*H
    float* den  = mx  + (long long)NNODE * HH;         // N*H
    float* hbuf = den + (long long)NNODE * HH;         // N*FF (A-operand overrun lands in agg)
    float* agg  = hbuf + (long long)NNODE * FFD;       // N*D  (last; never a GEMM A operand)
    float* srcD = xl;                                  // reuse xl after attention

    float* out_src   = (float*)d_out;                  // [N*D]
    float* out_eidx  = out_src + ND;                   // [2*E]
    float* out_alpha = out_eidx + 2LL * NEDGE;         // [E*H]

    const int T = 256;
    const int gND = (int)((ND + T - 1) / T);
    const int MB4 = (NNODE / 16 + 3) / 4;              // 782 M-blocks of 64 rows

    // 1. init accumulators
    k_init<<<gND, T, 0, stream>>>(scal, mx, den, agg, NNODE * HH, ND);
    // 2-3. graph LN 0
    k_reduce<<<2048, T, 0, stream>>>(x, ND, scal + 0, scal + 1);
    k_ln_apply<<<gND, T, 0, stream>>>(x, scal + 0, ln0_w, ln0_b, sA, ND, 1.0f / (float)ND);
    // 4-5. xl = sA@W_l+b_l ; xr = sA@W_r+b_r   (blocks = 782 * 512/128 = 3128)
    k_gemm_wmma<<<MB4 * (HC / 128), T, 0, stream>>>(sA, W_l, b_l, nullptr, xl, NNODE, DD, HC, 0);
    k_gemm_wmma<<<MB4 * (HC / 128), T, 0, stream>>>(sA, W_r, b_r, nullptr, xr, NNODE, DD, HC, 0);
    // 6. fused e-projection + logits + segment max (25000 edge tiles)
    k_edge_logits<<<25000 / 8, T, 0, stream>>>(eattr, W_e, att, eidx, xl, xr, lg, mx, 25000);
    // 7. exp + denom
    k_edge_exp<<<(NEDGE * HH + T - 1) / T, T, 0, stream>>>(eidx, lg, mx, den);
    // 8. alpha + head-reduced scatter aggregation (one wave per edge)
    k_edge_agg<<<(NEDGE * 32) / T, T, 0, stream>>>(lg, den, eidx, xl, agg, out_alpha);
    // 9. srcB = srcA + attnOut + bias_out
    k_res_bias<<<gND, T, 0, stream>>>(sA, bias_out, agg, ND);
    // 10-11. graph LN 1 -> srcC (into sA)
    k_reduce<<<2048, T, 0, stream>>>(agg, ND, scal + 2, scal + 3);
    k_ln_apply<<<gND, T, 0, stream>>>(agg, scal + 2, ln1_w, ln1_b, sA, ND, 1.0f / (float)ND);
    // 12. h = gelu(srcC@W1+b1)   (blocks = 782 * 2048/128 = 12512)
    k_gemm_wmma<<<MB4 * (FFD / 128), T, 0, stream>>>(sA, W1, b1, nullptr, hbuf, NNODE, DD, FFD, 1);
    // 13. srcD = h@W2 + b2 + srcC (blocks = 782 * 128/128 = 782)
    k_gemm_wmma<<<MB4 * (DD / 128), T, 0, stream>>>(hbuf, W2, b2, sA, srcD, NNODE, FFD, DD, 0);
    // 14-15. graph LN 2 -> d_out
    k_reduce<<<2048, T, 0, stream>>>(srcD, ND, scal + 4, scal + 5);
    k_ln_apply<<<gND, T, 0, stream>>>(srcD, scal + 4, ln2_w, ln2_b, out_src, ND, 1.0f / (float)ND);
    // 16. edge_index -> float output slot
    k_cast_idx<<<(2 * NEDGE + T - 1) / T, T, 0, stream>>>(eidx, out_eidx, 2 * NEDGE);
}